// SpanClassificationHead_635655159861
// MI455X (gfx1250) — compile-verified
//
#include <hip/hip_runtime.h>

typedef __attribute__((ext_vector_type(2))) float v2f;
typedef __attribute__((ext_vector_type(8))) float v8f;

// Problem constants (from reference setup_inputs)
constexpr int BB   = 2;       // batch
constexpr int S    = 384;     // seq len
constexpr int H    = 256;     // hidden
constexpr int WD   = 64;      // width-emb dim (H/4)
constexpr int R    = 576;     // 2H + WD
constexpr int NL   = 5;       // labels
constexpr int NSP  = 3785;    // number of spans
constexpr int NP   = 3840;    // padded spans (60 * 64)
constexpr int TQ   = NP / 64; // 60 tile-rows

// ---------------------------------------------------------------------------
// CDNA5 async global->LDS copy (ASYNCcnt-tracked DMA, no VGPR round trip).
// Builtin signature (from compiler diagnostic): param0 = int4-vector AS1*,
// param1 = int4-vector AS3*, then imm offset + imm cpol.
// Falls back to a synchronous generic-pointer copy if the builtin is absent.
// ---------------------------------------------------------------------------
#if defined(__has_builtin)
#if __has_builtin(__builtin_amdgcn_global_load_async_to_lds_b128)
#define ASYNC_LDS 1
#endif
#endif
#ifndef ASYNC_LDS
#define ASYNC_LDS 0
#endif

typedef int i4v __attribute__((ext_vector_type(4)));
typedef __attribute__((address_space(1))) i4v gas_i4;
typedef __attribute__((address_space(3))) i4v las_i4;

__device__ __forceinline__ void copy16(const float* __restrict__ g, v2f* l) {
#if ASYNC_LDS
  __builtin_amdgcn_global_load_async_to_lds_b128(
      (gas_i4*)(const_cast<float*>(g)), (las_i4*)l, 0, 0);
#else
  *reinterpret_cast<float4*>(l) = *reinterpret_cast<const float4*>(g);
#endif
}

template <int N>
__device__ __forceinline__ void wait_async() {
#if ASYNC_LDS
#if __has_builtin(__builtin_amdgcn_s_wait_asynccnt)
  __builtin_amdgcn_s_wait_asynccnt(N);
#else
  asm volatile("s_wait_asynccnt %0" ::"i"(N) : "memory");
#endif
#endif
}

// ---------------------------------------------------------------------------
// Kernel 1: build padded span representations  sp[b][n][r], zeros for n>=NSP
// ---------------------------------------------------------------------------
__global__ void build_spans(const float* __restrict__ hs,
                            const float* __restrict__ wemb,
                            float* __restrict__ sp) {
  int idx = blockIdx.x * blockDim.x + threadIdx.x;
  constexpr int total = BB * NP * R;
  if (idx >= total) return;
  int r = idx % R;
  int t = idx / R;
  int n = t % NP;
  int b = t / NP;
  float v = 0.0f;
  if (n < NSP) {
    int s, w;
    if (n < 3750) { s = n / 10; w = n - s * 10; }
    else {
      int m = n - 3750;
      s = 375; int cnt = 9;
      while (m >= cnt) { m -= cnt; cnt--; s++; }
      w = m;
    }
    int e = s + w;
    if (r < H)          v = hs[((size_t)b * S + s) * H + r];
    else if (r < 2 * H) v = hs[((size_t)b * S + e) * H + (r - H)];
    else                v = wemb[w * WD + (r - 2 * H)];
  }
  sp[idx] = v;
}

// ---------------------------------------------------------------------------
// Kernel 2: transpose bw (R x R) -> bwT so every GEMM is "B stored N x K"
// ---------------------------------------------------------------------------
__global__ void transpose_bw(const float* __restrict__ bw,
                             float* __restrict__ bwT) {
  int idx = blockIdx.x * blockDim.x + threadIdx.x;
  if (idx >= R * R) return;
  int k = idx % R;
  int n = idx / R;
  bwT[(size_t)n * R + k] = bw[(size_t)k * R + n];
}

// ---------------------------------------------------------------------------
// WMMA f32 GEMM core, C = A * B^T tile: A (M x K) row-major, Bm (N x K)
// row-major. 64x64 block tile, BK=16, 128 threads = 4 waves, each wave a
// 32x32 register tile (2x2 WMMA 16x16x4 accumulators).
// Both LDS tiles are K-pair-packed row-major [row][k/2] so a 16B global chunk
// X[row][k..k+3] is LDS-contiguous (async-DMA friendly) and each WMMA
// fragment is one aligned ds_load_b64.  Rows padded to 14 float2 (112B):
// 16B-aligned chunks, and 28*i mod 64 = 16 distinct multiples of 4 ->
// conflict-free b64 fragment reads for all 32 lanes (halves at +2 banks).
// ---------------------------------------------------------------------------
__device__ __forceinline__ void gemm_core_bt(const float* __restrict__ A, int lda,
                                             const float* __restrict__ Bm, int ldb,
                                             int K, int bm0, int bn0,
                                             v8f acc[2][2]) {
  __shared__ v2f As[2][64][14];
  __shared__ v2f Bs[2][64][14];

  const int tid  = threadIdx.x;      // 0..127
  const int lane = tid & 31;
  const int wave = tid >> 5;         // 0..3
  const int half = lane >> 4;        // 0/1
  const int l    = lane & 15;
  const int wm   = (wave >> 1) * 32; // wave row offset in block tile
  const int wn   = (wave & 1) * 32;  // wave col offset in block tile

  // Stage one 16-float-wide K-chunk of both tiles: 256 16B chunks per tile,
  // 128 threads x 2 -> each thread issues 2 A + 2 B copies (4 async ops).
  auto stage = [&](int kt, int p) {
#pragma unroll
    for (int u = 0; u < 2; ++u) {
      int q = tid + u * 128;          // 0..255
      int row = q >> 2, c2 = (q & 3) * 2;  // float2 index = (q&3)*4 / 2
      copy16(&A[(size_t)(bm0 + row) * lda + kt + (q & 3) * 4], &As[p][row][c2]);
      copy16(&Bm[(size_t)(bn0 + row) * ldb + kt + (q & 3) * 4], &Bs[p][row][c2]);
    }
  };

  auto compute = [&](int p) {
#pragma unroll
    for (int kk = 0; kk < 8; kk += 2) {
      const int kp = kk + half;       // K-pair index: lanes 16-31 take k+1 pair
      v2f a0 = As[p][wm + l][kp];
      v2f a1 = As[p][wm + 16 + l][kp];
      v2f b0 = Bs[p][wn + l][kp];
      v2f b1 = Bs[p][wn + 16 + l][kp];
      acc[0][0] = __builtin_amdgcn_wmma_f32_16x16x4_f32(false, a0, false, b0,
                                                        (short)0, acc[0][0], false, false);
      acc[0][1] = __builtin_amdgcn_wmma_f32_16x16x4_f32(false, a0, false, b1,
                                                        (short)0, acc[0][1], false, false);
      acc[1][0] = __builtin_amdgcn_wmma_f32_16x16x4_f32(false, a1, false, b0,
                                                        (short)0, acc[1][0], false, false);
      acc[1][1] = __builtin_amdgcn_wmma_f32_16x16x4_f32(false, a1, false, b1,
                                                        (short)0, acc[1][1], false, false);
    }
  };

  const int nk = K / 16;
  stage(0, 0);                           // issue chunk 0 -> buf 0
  for (int kt = 0; kt < nk; ++kt) {
    if (kt + 1 < nk) {
      stage((kt + 1) * 16, (kt + 1) & 1);  // DMA next chunk while we compute
      wait_async<4>();                     // chunk kt done (in-order), kt+1 flying
    } else {
      wait_async<0>();
    }
    __syncthreads();                       // chunk kt visible to all waves
    compute(kt & 1);
    __syncthreads();                       // safe to overwrite this buffer next
  }
}

// ---------------------------------------------------------------------------
// Generic GEMM kernel: C[b] = op(A[b] * B^T (+ bias)), exact-tiled
// ---------------------------------------------------------------------------
template <bool BIAS, bool RELU>
__global__ __launch_bounds__(128) void gemm_kernel(
    const float* __restrict__ A, int lda,
    const float* __restrict__ Bm, int ldb,
    const float* __restrict__ bias,
    float* __restrict__ C, int ldc, int K,
    int strideA, int strideC) {
  const int bm0 = blockIdx.x * 64;
  const int bn0 = blockIdx.y * 64;
  const int b   = blockIdx.z;
  A += (size_t)b * strideA;
  C += (size_t)b * strideC;

  v8f acc[2][2] = {};
  gemm_core_bt(A, lda, Bm, ldb, K, bm0, bn0, acc);

  const int lane = threadIdx.x & 31;
  const int wave = threadIdx.x >> 5;
  const int half = lane >> 4, l = lane & 15;
  const int wm = (wave >> 1) * 32, wn = (wave & 1) * 32;
#pragma unroll
  for (int mt = 0; mt < 2; ++mt)
#pragma unroll
    for (int nt = 0; nt < 2; ++nt)
#pragma unroll
      for (int r = 0; r < 8; ++r) {
        int i = bm0 + wm + mt * 16 + 8 * half + r;
        int j = bn0 + wn + nt * 16 + l;
        float v = acc[mt][nt][r];
        if (BIAS) v += bias[j];
        if (RELU) v = v > 0.0f ? v : 0.0f;
        C[(size_t)i * ldc + j] = v;
      }
}

// ---------------------------------------------------------------------------
// Link-score kernel: upper-triangular tiles of M = tmp * span^T, mirrored.
// ---------------------------------------------------------------------------
__global__ __launch_bounds__(128) void link_kernel(
    const float* __restrict__ tmp, const float* __restrict__ sp,
    const float* __restrict__ bbp, float* __restrict__ outL) {
  // map linear block id -> (ib, jb) with jb >= ib over TQ x TQ tiles
  int t = blockIdx.x;
  int ib = 0, rowlen = TQ;
  while (t >= rowlen) { t -= rowlen; ib++; rowlen--; }
  const int jb = ib + t;
  const int b  = blockIdx.z;

  const float* A  = tmp + (size_t)b * NP * R;
  const float* Bm = sp  + (size_t)b * NP * R;
  float* O        = outL + (size_t)b * NSP * NSP;
  const float bb  = bbp[0];

  v8f acc[2][2] = {};
  gemm_core_bt(A, R, Bm, R, R, ib * 64, jb * 64, acc);

  const int lane = threadIdx.x & 31;
  const int wave = threadIdx.x >> 5;
  const int half = lane >> 4, l = lane & 15;
  const int wm = (wave >> 1) * 32, wn = (wave & 1) * 32;
#pragma unroll
  for (int mt = 0; mt < 2; ++mt)
#pragma unroll
    for (int nt = 0; nt < 2; ++nt)
#pragma unroll
      for (int r = 0; r < 8; ++r) {
        int i = ib * 64 + wm + mt * 16 + 8 * half + r;
        int j = jb * 64 + wn + nt * 16 + l;
        if (i < NSP && j < NSP) {
          float v = acc[mt][nt][r] + bb;
          if (i < j) {
            O[(size_t)i * NSP + j] = v;
            O[(size_t)j * NSP + i] = v;
          } else if (i == j) {
            O[(size_t)i * NSP + j] = 0.0f;
          } // i > j inside diagonal tiles: mirrored by the (j,i) lane
        }
      }
}

// ---------------------------------------------------------------------------
// Span logits: out[b][n][l] = dot(h[b][n], w2[l]) + b2[l]   (tiny, VALU)
// ---------------------------------------------------------------------------
__global__ void logits_kernel(const float* __restrict__ h,
                              const float* __restrict__ w2,
                              const float* __restrict__ b2,
                              float* __restrict__ out) {
  int idx = blockIdx.x * blockDim.x + threadIdx.x;
  constexpr int total = BB * NSP * NL;
  if (idx >= total) return;
  int l = idx % NL;
  int t = idx / NL;
  int n = t % NSP;
  int b = t / NSP;
  const float* hr = h + ((size_t)b * NP + n) * H;
  const float* wr = w2 + (size_t)l * H;
  float s = 0.0f;
#pragma unroll 4
  for (int k = 0; k < H; ++k) s += hr[k] * wr[k];
  out[idx] = s + b2[l];
}

// ---------------------------------------------------------------------------
extern "C" void kernel_launch(void* const* d_in, const int* in_sizes, int n_in,
                              void* d_out, int out_size, void* d_ws, size_t ws_size,
                              hipStream_t stream) {
  const float* hs   = (const float*)d_in[0];
  const float* wemb = (const float*)d_in[1];
  const float* w1   = (const float*)d_in[2];  // (H, R)  == N x K, ready for B^T
  const float* b1   = (const float*)d_in[3];  // (H)
  const float* w2   = (const float*)d_in[4];  // (NL, H)
  const float* b2   = (const float*)d_in[5];  // (NL)
  const float* bw   = (const float*)d_in[6];  // (1, R, R)
  const float* bb   = (const float*)d_in[7];  // (1)
  float* out = (float*)d_out;
  float* ws  = (float*)d_ws;

  float* sp  = ws;                              // BB * NP * R
  float* tmp = sp  + (size_t)BB * NP * R;       // BB * NP * R
  float* h   = tmp + (size_t)BB * NP * R;       // BB * NP * H
  float* bwT = h   + (size_t)BB * NP * H;       // R * R

  // 1) gather padded span representations + transpose bw
  {
    int total = BB * NP * R;
    build_spans<<<(total + 255) / 256, 256, 0, stream>>>(hs, wemb, sp);
    transpose_bw<<<(R * R + 255) / 256, 256, 0, stream>>>(bw, bwT);
  }
  // 2) h = relu(sp * w1^T + b1)   : (NP x R) * (R x H)
  gemm_kernel<true, true>
      <<<dim3(NP / 64, H / 64, BB), 128, 0, stream>>>(
          sp, R, w1, R, b1, h, H, R, NP * R, NP * H);
  // 3) tmp = sp * bw = sp * (bwT)^T : (NP x R) * (R x R)
  gemm_kernel<false, false>
      <<<dim3(NP / 64, R / 64, BB), 128, 0, stream>>>(
          sp, R, bwT, R, nullptr, tmp, R, R, NP * R, NP * R);
  // 4) span logits (output part 1)
  {
    int total = BB * NSP * NL;
    logits_kernel<<<(total + 255) / 256, 256, 0, stream>>>(h, w2, b2, out);
  }
  // 5) link scores (output part 2): upper-tri tiles of tmp * sp^T, mirrored
  link_kernel<<<dim3(TQ * (TQ + 1) / 2, 1, BB), 128, 0, stream>>>(
      tmp, sp, bb, out + (size_t)BB * NSP * NL);
}